// AttentionNeNode_30554397343922
// MI455X (gfx1250) — compile-verified
//
#include <hip/hip_runtime.h>

typedef float v2f __attribute__((ext_vector_type(2)));
typedef float v8f __attribute__((ext_vector_type(8)));

static constexpr int R_DIM  = 8192;
static constexpr int F_DIM  = 4096;
static constexpr int L_DIM  = 2048;
static constexpr int NPAD   = 16;          // pad the 3 output cols to 16 for WMMA
static constexpr int KSPLIT = 4;
static constexpr int KCHUNK = F_DIM / KSPLIT; // 1024

// -----------------------------------------------------------------------------
// Phase 1: densify the column-gather, stored TRANSPOSED: w_t[j][c] (16 x 4096),
// w_t[j][c] = sum over l of weights[l][j] where in_idxs[l]==c  (j<3; rows 3..15
// are zero padding).  Transposed layout makes the WMMA B-fragment loads a
// single contiguous b64 per lane.  Deterministic (no atomics): one thread per
// (c, j) scans the 2048 indices (L0-resident, trivial).
// -----------------------------------------------------------------------------
__global__ void build_wdense(const int* __restrict__ idxs,
                             const float* __restrict__ wts,
                             float* __restrict__ wt) {
  int t = blockIdx.x * blockDim.x + threadIdx.x;   // F_DIM*NPAD = 65536 threads
  int c = t >> 4;
  int j = t & 15;
  float acc = 0.f;
  if (j < 3) {
    for (int l = 0; l < L_DIM; ++l) {
      if (idxs[l] == c) acc += wts[l * 3 + j];
    }
  }
  wt[(size_t)j * F_DIM + c] = acc;                 // transposed store
}

// -----------------------------------------------------------------------------
// Phase 2: qkv = inputs(8192x4096) @ w_dense(4096x16) via V_WMMA_F32_16X16X4_F32.
// One wave per (16-row tile, K-slice).  A-layout (ISA 7.12.2, 16x4 f32):
//   lanes 0-15: M=lane,  v0=K0 v1=K1 ; lanes 16-31: M=lane-16, v0=K2 v1=K3
// -> each lane streams one aligned float2 from its input row per step.
// B (4x16) mirrors it; with w_t[n][k] both B elements a lane needs
// ((K,N=l16),(K+1,N=l16)) are adjacent -> one aligned float2 load per step.
// C-layout: lanes 0-15 -> N=lane, M=vgpr 0..7 ; lanes 16-31 -> N=lane-16, M=8+vgpr.
// K split 4 ways into separate partial buffers (summed later, deterministic).
// -----------------------------------------------------------------------------
__global__ void qkv_gemm_wmma(const float* __restrict__ inp,
                              const float* __restrict__ wt,
                              float* __restrict__ qkv_part) {
  int gw   = (blockIdx.x * blockDim.x + threadIdx.x) >> 5; // global wave id
  int lane = threadIdx.x & 31;
  int tile = gw >> 2;          // 0..511  (16-row tile)
  int ks   = gw & 3;           // K slice
  int half = lane >> 4;        // 0 or 1
  int l16  = lane & 15;
  int row  = tile * 16 + l16;

  const float* a_ptr = inp + (size_t)row * F_DIM + (half << 1);
  const float* b_ptr = wt + (size_t)l16 * F_DIM + (half << 1);

  int k0 = ks * KCHUNK;
  v8f c = {};
#pragma unroll 4
  for (int kk = k0; kk < k0 + KCHUNK; kk += 4) {
    v2f a = *reinterpret_cast<const v2f*>(a_ptr + kk);  // A: (K, K+1) of row
    v2f b = *reinterpret_cast<const v2f*>(b_ptr + kk);  // B: (K, K+1) of col l16
    c = __builtin_amdgcn_wmma_f32_16x16x4_f32(false, a, false, b,
                                              (short)0, c, false, false);
  }

  // Only output columns 0..2 matter (q,k,v).  Stride-4 layout for alignment.
  float* dst = qkv_part + (size_t)ks * (R_DIM * 4);
  if (l16 < 3) {
    int mbase = tile * 16 + half * 8;
#pragma unroll
    for (int m = 0; m < 8; ++m)
      dst[(size_t)(mbase + m) * 4 + l16] = c[m];
  }
}

// -----------------------------------------------------------------------------
// Phase 3: out = sigmoid( softmax_r(q_last * k_r) . v ) over 8192 rows.
// Single block; sums the K-split partials in a fixed order (deterministic).
// -----------------------------------------------------------------------------
__global__ void attn_reduce(const float* __restrict__ qkv_part,
                            float* __restrict__ out) {
  __shared__ float sm[1024];
  int tid = threadIdx.x;

  auto elem = [&](int r, int n) {
    float s = 0.f;
#pragma unroll
    for (int p = 0; p < KSPLIT; ++p)
      s += qkv_part[(size_t)p * (R_DIM * 4) + (size_t)r * 4 + n];
    return s;
  };

  float qlast = elem(R_DIM - 1, 0);

  float pv[8], vv[8];
  float lmax = -3.402823466e38f;
#pragma unroll
  for (int i = 0; i < 8; ++i) {
    int r = tid * 8 + i;
    pv[i] = qlast * elem(r, 1);   // logits (d_k = 1 -> scale = 1)
    vv[i] = elem(r, 2);
    lmax = fmaxf(lmax, pv[i]);
  }

  sm[tid] = lmax;
  __syncthreads();
  for (int s = 512; s > 0; s >>= 1) {
    if (tid < s) sm[tid] = fmaxf(sm[tid], sm[tid + s]);
    __syncthreads();
  }
  float m = sm[0];
  __syncthreads();

  float den = 0.f, num = 0.f;
#pragma unroll
  for (int i = 0; i < 8; ++i) {
    float e = __expf(pv[i] - m);
    den += e;
    num += e * vv[i];
  }

  sm[tid] = den;
  __syncthreads();
  for (int s = 512; s > 0; s >>= 1) {
    if (tid < s) sm[tid] += sm[tid + s];
    __syncthreads();
  }
  den = sm[0];
  __syncthreads();

  sm[tid] = num;
  __syncthreads();
  for (int s = 512; s > 0; s >>= 1) {
    if (tid < s) sm[tid] += sm[tid + s];
    __syncthreads();
  }
  num = sm[0];

  if (tid == 0) {
    float x = num / den;
    out[0] = 1.f / (1.f + __expf(-x));
  }
}

// -----------------------------------------------------------------------------
extern "C" void kernel_launch(void* const* d_in, const int* in_sizes, int n_in,
                              void* d_out, int out_size, void* d_ws, size_t ws_size,
                              hipStream_t stream) {
  (void)in_sizes; (void)n_in; (void)out_size; (void)ws_size;
  const float* inputs = (const float*)d_in[0];
  const int*   idxs   = (const int*)d_in[1];
  const float* wts    = (const float*)d_in[2];

  float* ws     = (float*)d_ws;
  float* wdense = ws;                      // NPAD*F_DIM floats (transposed, 256 KB)
  float* qkvp   = ws + NPAD * F_DIM;       // KSPLIT*R_DIM*4 floats        (512 KB)

  build_wdense<<<(F_DIM * NPAD) / 256, 256, 0, stream>>>(idxs, wts, wdense);

  int waves = (R_DIM / 16) * KSPLIT;       // 2048 waves
  qkv_gemm_wmma<<<waves * 32 / 256, 256, 0, stream>>>(inputs, wdense, qkvp);

  attn_reduce<<<1, 1024, 0, stream>>>(qkvp, (float*)d_out);
}